// Model_FEVER_17197049053894
// MI455X (gfx1250) — compile-verified
//
#include <hip/hip_runtime.h>
#include <math.h>

#define GAT_H 12
#define GAT_D 64

typedef __attribute__((ext_vector_type(2))) float v2f;
typedef __attribute__((ext_vector_type(8))) float v8f;

// ---- order-preserving float<->uint mapping for atomic float max via atomicMax(u32) ----
__device__ __forceinline__ unsigned f2ord(float f) {
    unsigned b = __float_as_uint(f);
    return b ^ ((b >> 31) ? 0xFFFFFFFFu : 0x80000000u);
}
__device__ __forceinline__ float ord2f(unsigned u) {
    return __uint_as_float((u >> 31) ? (u ^ 0x80000000u) : ~u);
}

// =====================================================================================
// WMMA GEMM: Y[R,64] = X[R,64] @ W[64,64]^T  using V_WMMA_F32_16X16X4_F32.
// One wave computes a 16-row x 64-col strip (4 accumulator tiles), 16 K-steps of 4.
// If addIn != nullptr: Y = relu(gemm + addIn)  (fused residual epilogue).
// ISA f32 layouts (wave32):
//   A 16x4:  lane L(0..15): VGPR{0,1}=A[M=L, K={0,1}];  lane L(16..31): K={2,3}
//   B 4x16:  lanes 0..15 hold K={0,1}, lanes 16..31 hold K={2,3}; N = lane%16
//   C/D 16x16: VGPR r: M = r + 8*(lane>=16), N = lane%16
// =====================================================================================
__global__ void gat_gemm64(const float* __restrict__ X,
                           const float* __restrict__ W,
                           float* __restrict__ Y,
                           const float* __restrict__ addIn,
                           int R) {
    __shared__ float lw[64 * 66];   // W staged in LDS, stride 66 to avoid bank conflicts
    for (int i = threadIdx.x; i < 64 * 64; i += blockDim.x)
        lw[(i >> 6) * 66 + (i & 63)] = W[i];
    __syncthreads();

    const int lane  = threadIdx.x & 31;
    const int wave  = threadIdx.x >> 5;
    const int mtile = blockIdx.x * 8 + wave;
    const int rtiles = R >> 4;
    if (mtile >= rtiles) return;          // whole-wave exit; WMMA waves keep EXEC all-1s

    const int half = lane >> 4;           // 0: lanes 0-15, 1: lanes 16-31
    const int l16  = lane & 15;
    const float* arow = X + ((size_t)(mtile * 16 + l16)) * 64 + 2 * half;

    v8f acc[4] = {};
#pragma unroll
    for (int kb = 0; kb < 16; ++kb) {
        const int k0 = kb * 4;
        v2f a = *(const v2f*)(arow + k0);
#pragma unroll
        for (int c = 0; c < 4; ++c) {
            v2f b = *(const v2f*)(&lw[(c * 16 + l16) * 66 + k0 + 2 * half]);
            acc[c] = __builtin_amdgcn_wmma_f32_16x16x4_f32(
                false, a, false, b, (short)0, acc[c], false, false);
        }
    }

#pragma unroll
    for (int c = 0; c < 4; ++c) {
#pragma unroll
        for (int r = 0; r < 8; ++r) {
            const int row = mtile * 16 + r + 8 * half;
            const int col = c * 16 + l16;
            const size_t idx = (size_t)row * 64 + col;
            float v = acc[c][r];
            if (addIn) {
                v += addIn[idx];
                v = v > 0.0f ? v : 0.0f;
            }
            Y[idx] = v;
        }
    }
}

// a1[t] = <ft[t,:], attn_l[h,:]>, a2[t] = <ft[t,:], attn_r[h,:]>, t = n*H + h
__global__ void gat_a1a2(const float* __restrict__ ft,
                         const float* __restrict__ attn_l,
                         const float* __restrict__ attn_r,
                         float* __restrict__ a1, float* __restrict__ a2, int NH) {
    int t = blockIdx.x * blockDim.x + threadIdx.x;
    if (t >= NH) return;
    int h = t % GAT_H;
    const float4* f  = (const float4*)(ft + (size_t)t * 64);
    const float4* al = (const float4*)(attn_l + h * 64);
    const float4* ar = (const float4*)(attn_r + h * 64);
    float s1 = 0.f, s2 = 0.f;
#pragma unroll 4
    for (int i = 0; i < 16; ++i) {
        float4 fv = f[i], lv = al[i], rv = ar[i];
        s1 += fv.x * lv.x + fv.y * lv.y + fv.z * lv.z + fv.w * lv.w;
        s2 += fv.x * rv.x + fv.y * rv.y + fv.z * rv.z + fv.w * rv.w;
    }
    a1[t] = s1;
    a2[t] = s2;
}

__device__ __forceinline__ float edge_logit(const float* a1, const float* a2,
                                            int s, int d, int h) {
    float x = a1[s * GAT_H + h] + a2[d * GAT_H + h];
    return x > 0.f ? x : 0.2f * x;   // leaky_relu 0.2
}

// pass 1: segment max over dst via ordered-uint atomicMax
__global__ void gat_edge_max(const int* __restrict__ src, const int* __restrict__ dst,
                             const float* __restrict__ a1, const float* __restrict__ a2,
                             unsigned* __restrict__ mord, int EH) {
    int t = blockIdx.x * blockDim.x + threadIdx.x;
    if (t >= EH) return;
    int e = t / GAT_H, h = t - e * GAT_H;
    int s = src[e], d = dst[e];
    float v = edge_logit(a1, a2, s, d, h);
    atomicMax(&mord[d * GAT_H + h], f2ord(v));
}

// pass 2: ee = exp(e - m[dst]); ebuf <- ee; denom[dst] += ee
__global__ void gat_edge_expsum(const int* __restrict__ src, const int* __restrict__ dst,
                                const float* __restrict__ a1, const float* __restrict__ a2,
                                const unsigned* __restrict__ mord,
                                float* __restrict__ ebuf, float* __restrict__ denom, int EH) {
    int t = blockIdx.x * blockDim.x + threadIdx.x;
    if (t >= EH) return;
    int e = t / GAT_H, h = t - e * GAT_H;
    int s = src[e], d = dst[e];
    float v  = edge_logit(a1, a2, s, d, h);
    float mx = ord2f(mord[d * GAT_H + h]);
    float ee = __expf(v - mx);
    ebuf[t] = ee;
    atomicAdd(&denom[d * GAT_H + h], ee);
}

// pass 3: agg[dst,h,:] += (ebuf/denom[dst,h]) * ft[src,h,:]; thread = (edge, head, 4-float chunk)
__global__ void gat_scatter(const int* __restrict__ src, const int* __restrict__ dst,
                            const float* __restrict__ ebuf, const float* __restrict__ denom,
                            const float* __restrict__ ft, float* __restrict__ agg, int EH16) {
    int t = blockIdx.x * blockDim.x + threadIdx.x;
    if (t >= EH16) return;
    int c   = t & 15;
    int rem = t >> 4;
    int h = rem % GAT_H;
    int e = rem / GAT_H;
    int s = src[e], d = dst[e];
    float alpha = ebuf[rem] / denom[d * GAT_H + h];
    float4 fv = *(const float4*)(ft + ((size_t)(s * GAT_H + h)) * 64 + c * 4);
    float* ap = agg + ((size_t)(d * GAT_H + h)) * 64 + c * 4;
    atomicAdd(ap + 0, alpha * fv.x);
    atomicAdd(ap + 1, alpha * fv.y);
    atomicAdd(ap + 2, alpha * fv.z);
    atomicAdd(ap + 3, alpha * fv.w);
}

extern "C" void kernel_launch(void* const* d_in, const int* in_sizes, int n_in,
                              void* d_out, int out_size, void* d_ws, size_t ws_size,
                              hipStream_t stream) {
    const float* h_in   = (const float*)d_in[0];   // [N, H*D]
    const float* W_fc   = (const float*)d_in[1];   // [64,64]
    const float* attn_l = (const float*)d_in[2];   // [H,64]
    const float* attn_r = (const float*)d_in[3];   // [H,64]
    const float* W_res  = (const float*)d_in[4];   // [64,64]
    const int*   src    = (const int*)d_in[5];     // [E]
    const int*   dst    = (const int*)d_in[6];     // [E]

    const int NH = in_sizes[0] / GAT_D;            // N*H rows of the 64-wide GEMMs
    const int E  = in_sizes[5];
    const int EH = E * GAT_H;

    // workspace layout (floats)
    float*    ws    = (float*)d_ws;
    float*    ft    = ws;                          // NH*64
    float*    agg   = ft  + (size_t)NH * 64;       // NH*64
    float*    a1    = agg + (size_t)NH * 64;       // NH
    float*    a2    = a1  + NH;                    // NH
    unsigned* mord  = (unsigned*)(a2 + NH);        // NH
    float*    denom = (float*)(mord + NH);         // NH
    float*    ebuf  = denom + NH;                  // E*H

    // zero agg..denom in one contiguous memset (a1/a2 are fully overwritten anyway)
    hipMemsetAsync(agg, 0, ((size_t)NH * 64 + 4 * (size_t)NH) * sizeof(float), stream);

    const int rtiles = NH >> 4;                    // 16-row WMMA strips
    dim3 gemm_grid((rtiles + 7) / 8);

    // ft = X @ W_fc^T   (fp32 WMMA)
    gat_gemm64<<<gemm_grid, 256, 0, stream>>>(h_in, W_fc, ft, nullptr, NH);
    // a1, a2
    gat_a1a2<<<(NH + 255) / 256, 256, 0, stream>>>(ft, attn_l, attn_r, a1, a2, NH);
    // segment softmax over incoming edges
    gat_edge_max   <<<(EH + 255) / 256, 256, 0, stream>>>(src, dst, a1, a2, mord, EH);
    gat_edge_expsum<<<(EH + 255) / 256, 256, 0, stream>>>(src, dst, a1, a2, mord, ebuf, denom, EH);
    // weighted scatter-sum of messages
    gat_scatter<<<(EH * 16 + 255) / 256, 256, 0, stream>>>(src, dst, ebuf, denom, ft, agg, EH * 16);
    // out = relu(X @ W_res^T + agg)   (fp32 WMMA, fused epilogue)
    gat_gemm64<<<gemm_grid, 256, 0, stream>>>(h_in, W_res, (float*)d_out, agg, NH);
}